// DirectionalGraphConvolution_46789373723027
// MI455X (gfx1250) — compile-verified
//
#include <hip/hip_runtime.h>
#include <hip/hip_bf16.h>

typedef __attribute__((ext_vector_type(2))) float v2f;
typedef __attribute__((ext_vector_type(8))) float v8f;

#define IN_C 128
#define OUT_C 128

// ---------------------------------------------------------------------------
// 0) init: zero the output accumulator (float4), set deg[i] = 1.0 (self-loop)
// ---------------------------------------------------------------------------
__global__ void gcn_init_kernel(float4* __restrict__ out4, float* __restrict__ deg,
                                int n_out4, int n_nodes) {
    int i = blockIdx.x * blockDim.x + threadIdx.x;
    if (i < n_out4) out4[i] = make_float4(0.f, 0.f, 0.f, 0.f);
    if (i < n_nodes) deg[i] = 1.0f;
}

// ---------------------------------------------------------------------------
// 1) GEMM: xw[N,128] = x[N,128] @ W[128,128] via V_WMMA_F32_16X16X4_F32.
//    Block = 256 threads = 8 waves. Each wave owns one 16x16 output tile:
//    rows  = blockIdx.x*16 .. +15, cols = wave*16 .. +15.
//    W staged in LDS (64 KB; WGP has 320 KB).
//    A layout (16x4 f32, 2 VGPRs): lanes 0-15 -> K=0,1 ; lanes 16-31 -> K=2,3.
//    B layout (4x16 f32, 2 VGPRs): VGPR0 = rows K=0 (lanes 0-15) / K=2 (16-31),
//                                  VGPR1 = rows K=1 / K=3.
// ---------------------------------------------------------------------------
__global__ void __launch_bounds__(256)
gcn_gemm_kernel(const float* __restrict__ x, const float* __restrict__ w,
                float* __restrict__ xw, int n_nodes) {
    __shared__ float Wl[IN_C * OUT_C];   // 64 KB

    // cooperative load of W into LDS (float4 stripes)
    {
        const float4* w4  = (const float4*)w;
        float4*       wl4 = (float4*)Wl;
        for (int i = threadIdx.x; i < (IN_C * OUT_C) / 4; i += blockDim.x)
            wl4[i] = w4[i];
    }
    __syncthreads();

    const int lane = threadIdx.x & 31;
    const int wv   = threadIdx.x >> 5;
    const int half = lane >> 4;       // 0: K pair {0,1}, 1: K pair {2,3}
    const int l    = lane & 15;
    const int row0 = blockIdx.x * 16;
    const int col0 = wv * 16;

    const float* arow = x + (size_t)(row0 + l) * IN_C + half * 2;

    v8f c = {};
    #pragma unroll 4
    for (int kb = 0; kb < IN_C; kb += 4) {
        v2f a = *(const v2f*)(arow + kb);                       // 8B-aligned
        v2f b;
        b.x = Wl[(kb + half * 2 + 0) * OUT_C + col0 + l];
        b.y = Wl[(kb + half * 2 + 1) * OUT_C + col0 + l];
        c = __builtin_amdgcn_wmma_f32_16x16x4_f32(
                /*neg_a=*/false, a, /*neg_b=*/false, b,
                /*c_mod=*/(short)0, c, /*reuse_a=*/false, /*reuse_b=*/false);
    }

    // D layout: VGPR r -> M = r (lanes 0-15) / M = 8+r (lanes 16-31), N = l
    float* orow = xw + (size_t)(row0 + half * 8) * OUT_C + col0 + l;
    if (row0 + 16 <= n_nodes) {
        // full tile in bounds: single uniform branch, clause of 8 stores
        #pragma unroll
        for (int r = 0; r < 8; ++r) orow[(size_t)r * OUT_C] = c[r];
    } else {
        #pragma unroll
        for (int r = 0; r < 8; ++r) {
            int row = row0 + half * 8 + r;
            if (row < n_nodes) orow[(size_t)r * OUT_C] = c[r];
        }
    }
}

// ---------------------------------------------------------------------------
// 2) degree accumulation: deg[dst] += w  (deg pre-initialized to 1.0)
// ---------------------------------------------------------------------------
__global__ void gcn_deg_kernel(const long long* __restrict__ ei,
                               const float* __restrict__ ew,
                               float* __restrict__ deg, int n_edges) {
    int e = blockIdx.x * blockDim.x + threadIdx.x;
    if (e < n_edges) {
        int d = (int)ei[(size_t)n_edges + e];   // dst row of edge_index
        __hip_atomic_fetch_add(&deg[d], ew[e],
                               __ATOMIC_RELAXED, __HIP_MEMORY_SCOPE_AGENT);
    }
}

// ---------------------------------------------------------------------------
// 3) dinv = deg > 0 ? rsqrt(deg) : 0   (in place)
// ---------------------------------------------------------------------------
__global__ void gcn_dinv_kernel(float* __restrict__ deg, int n_nodes) {
    int i = blockIdx.x * blockDim.x + threadIdx.x;
    if (i < n_nodes) {
        float v = deg[i];
        deg[i] = (v > 0.0f) ? rsqrtf(v) : 0.0f;
    }
}

// ---------------------------------------------------------------------------
// 4) edge scatter: one wave32 per edge. Lane 0 computes
//    norm = dinv[src]*w*dinv[dst], broadcast; each lane moves a float4 of the
//    128-channel row: load xw[src], scale, fp32-atomic-add into out[dst].
// ---------------------------------------------------------------------------
__global__ void __launch_bounds__(256)
gcn_scatter_kernel(const long long* __restrict__ ei,
                   const float* __restrict__ ew,
                   const float* __restrict__ dinv,
                   const float* __restrict__ xw,
                   float* __restrict__ out, int n_edges) {
    int gid  = blockIdx.x * blockDim.x + threadIdx.x;
    int e    = gid >> 5;
    int lane = gid & 31;
    if (e >= n_edges) return;

    int s = 0, d = 0;
    float norm = 0.0f;
    if (lane == 0) {
        s = (int)ei[e];
        d = (int)ei[(size_t)n_edges + e];
        norm = dinv[s] * ew[e] * dinv[d];
    }
    s    = __shfl(s, 0, 32);
    d    = __shfl(d, 0, 32);
    norm = __shfl(norm, 0, 32);

    int c = lane * 4;
    float4 v = *(const float4*)(xw + (size_t)s * OUT_C + c);
    float* o = out + (size_t)d * OUT_C + c;
    __hip_atomic_fetch_add(o + 0, v.x * norm, __ATOMIC_RELAXED, __HIP_MEMORY_SCOPE_AGENT);
    __hip_atomic_fetch_add(o + 1, v.y * norm, __ATOMIC_RELAXED, __HIP_MEMORY_SCOPE_AGENT);
    __hip_atomic_fetch_add(o + 2, v.z * norm, __ATOMIC_RELAXED, __HIP_MEMORY_SCOPE_AGENT);
    __hip_atomic_fetch_add(o + 3, v.w * norm, __ATOMIC_RELAXED, __HIP_MEMORY_SCOPE_AGENT);
}

// ---------------------------------------------------------------------------
// 5) finalize: out = relu(out + xw*dinv^2 (self-loop) + bias), float4, in place
// ---------------------------------------------------------------------------
__global__ void gcn_finalize_kernel(float4* __restrict__ out4,
                                    const float4* __restrict__ xw4,
                                    const float* __restrict__ dinv,
                                    const float4* __restrict__ bias4, int total4) {
    int i = blockIdx.x * blockDim.x + threadIdx.x;
    if (i < total4) {
        int row = i >> 5;            // (i*4) / 128
        int c4  = i & 31;            // float4 index within row
        float di = dinv[row];
        float s  = di * di;
        float4 a = out4[i];
        float4 x = xw4[i];
        float4 b = bias4[c4];
        float4 r;
        r.x = fmaxf(a.x + x.x * s + b.x, 0.0f);
        r.y = fmaxf(a.y + x.y * s + b.y, 0.0f);
        r.z = fmaxf(a.z + x.z * s + b.z, 0.0f);
        r.w = fmaxf(a.w + x.w * s + b.w, 0.0f);
        out4[i] = r;
    }
}

extern "C" void kernel_launch(void* const* d_in, const int* in_sizes, int n_in,
                              void* d_out, int out_size, void* d_ws, size_t ws_size,
                              hipStream_t stream) {
    const float*     x    = (const float*)d_in[0];
    const long long* ei   = (const long long*)d_in[1];   // int64 [2, E]
    const float*     ew   = (const float*)d_in[2];
    const float*     w    = (const float*)d_in[3];
    const float*     bias = (const float*)d_in[4];
    float*           out  = (float*)d_out;

    const int n_nodes = in_sizes[0] / IN_C;    // 50000
    const int n_edges = in_sizes[2];           // 800000
    const int total   = n_nodes * OUT_C;
    const int total4  = total / 4;

    // workspace layout: xw [N*128 f32] | deg/dinv [N f32]
    float* xw  = (float*)d_ws;
    float* deg = xw + (size_t)n_nodes * OUT_C;

    // 0) init accumulators (float4 zero fill of out; deg = 1.0)
    gcn_init_kernel<<<(total4 + 255) / 256, 256, 0, stream>>>(
        (float4*)out, deg, total4, n_nodes);
    // 1) WMMA fp32 GEMM
    gcn_gemm_kernel<<<(n_nodes + 15) / 16, 256, 0, stream>>>(x, w, xw, n_nodes);
    // 2) degree
    gcn_deg_kernel<<<(n_edges + 255) / 256, 256, 0, stream>>>(ei, ew, deg, n_edges);
    // 3) rsqrt
    gcn_dinv_kernel<<<(n_nodes + 255) / 256, 256, 0, stream>>>(deg, n_nodes);
    // 4) edge gather-scale-scatter (one wave per edge)
    gcn_scatter_kernel<<<(n_edges + 7) / 8, 256, 0, stream>>>(ei, ew, deg, xw, out, n_edges);
    // 5) self-loop + bias + relu (float4)
    gcn_finalize_kernel<<<(total4 + 255) / 256, 256, 0, stream>>>(
        (float4*)out, (const float4*)xw, deg, (const float4*)bias, total4);
}